// CGLayers_74363063763117
// MI455X (gfx1250) — compile-verified
//
#include <hip/hip_runtime.h>
#include <hip/hip_bf16.h>
#include <math.h>

typedef __attribute__((ext_vector_type(2))) float v2f;
typedef __attribute__((ext_vector_type(4))) float v4f;
typedef __attribute__((ext_vector_type(8))) float v8f;

// ---------------- constants ----------------
// B=2, A=32, C=16, MAXL=2.  Coupling blocks in reference traversal order:
__device__ const int BLK_L1[15] = {0,0,0,1,1,1,1,1,1,2,2,2,2,2,2};
__device__ const int BLK_L2[15] = {0,1,2,0,1,1,1,2,2,0,1,1,2,2,2};
__device__ const int LOUT_BLK[3][6] = {{0,4,12,0,0,0},{1,3,5,7,10,13},{2,6,8,9,11,14}};
__device__ const int LOUT_NBLK_D[3] = {3,6,6};
__device__ const int ROWBASE_D[3]   = {0,1,4};          // row base of l-part in 9-row layout
__device__ const size_t REGC_D[3]   = {0,393216,1179648}; // cgnl fragment regions

__device__ __forceinline__ int rowb(int l) { return l==0 ? 0 : (l==1 ? 1 : 4); }

// ---------------- workspace layout (floats) ----------------
#define OFF_CG      ((size_t)0)                       // 15*128 = 1920 (pad 2048)
#define OFF_PARTS_A ((size_t)2048)                    // 2*32*9*16 = 9216
#define OFF_PARTS_B (OFF_PARTS_A + 9216)
#define OFF_MNL     (OFF_PARTS_B + 9216)              // 9216
#define OFF_CGNL    (OFF_MNL + 9216)                  // 32 a * 16 rows * 3840 tau = 1966080
#define OFF_GS      (OFF_CGNL + 1966080)              // 3932160 (frag-swizzled G)
#define OFF_WD      (OFF_GS + 3932160)                // 2 * 3932160
#define WD_LAYER    ((size_t)3932160)
// within-layer Wd / Gs per-l offsets: {0, 786432, 2359296}

// ---------------- CG coefficients (computed on device, graph-safe) ----------------
__device__ double dfact_d(int n){ double r=1.0; for(int i=2;i<=n;++i) r*=(double)i; return r; }

__device__ float cg_coeff_d(int l1,int l2,int l,int m1,int m2,int m){
  if (m1+m2 != m) return 0.f;
  if (l < abs(l1-l2) || l > l1+l2) return 0.f;
  double pre = sqrt((double)(2*l+1)*dfact_d(l+l1-l2)*dfact_d(l-l1+l2)*dfact_d(l1+l2-l)/dfact_d(l1+l2+l+1));
  pre *= sqrt(dfact_d(l+m)*dfact_d(l-m)*dfact_d(l1+m1)*dfact_d(l1-m1)*dfact_d(l2+m2)*dfact_d(l2-m2));
  double s = 0.0;
  for (int k=0;k<=l1+l2;++k){
    int d0=k, d1=l1+l2-l-k, d2=l1-m1-k, d3=l2+m2-k, d4=l-l2+m1+k, d5=l-l1-m2+k;
    if (d0<0||d1<0||d2<0||d3<0||d4<0||d5<0) continue;
    double term = 1.0/(dfact_d(d0)*dfact_d(d1)*dfact_d(d2)*dfact_d(d3)*dfact_d(d4)*dfact_d(d5));
    s += (k&1) ? -term : term;
  }
  return (float)(pre*s);
}

// CG table: blk*128 + (x*n2 + y)*nl + z
__global__ void k_cg_init(float* __restrict__ cg){
  int blk = blockIdx.x, e = threadIdx.x;            // <<<15,128>>>
  int l1=BLK_L1[blk], l2=BLK_L2[blk];
  int lo = -1;
  for (int L=0; L<3; ++L) for (int s=0; s<LOUT_NBLK_D[L]; ++s) if (LOUT_BLK[L][s]==blk) lo=L;
  int n1=2*l1+1, n2=2*l2+1, nl=2*lo+1;
  int prod = n1*n2*nl;
  float v = 0.f;
  if (e < prod){
    int x = e/(n2*nl), y = (e/nl)%n2, z = e%nl;
    v = cg_coeff_d(l1,l2,lo, x-l1, y-l2, z-lo);
  }
  cg[blk*128 + e] = v;
}

// ---------------- pack inputs into [b][a][9][16] ----------------
__global__ void k_pack(const float* __restrict__ v0, const float* __restrict__ v1,
                       const float* __restrict__ v2, float* __restrict__ parts){
  int idx = blockIdx.x*256 + threadIdx.x;           // 9216 total
  if (idx >= 9216) return;
  int c = idx & 15, row = (idx>>4)%9, ba = idx/144;
  float v;
  if (row == 0)      v = v0[(size_t)ba*16 + c];
  else if (row < 4)  v = v1[((size_t)ba*3 + (row-1))*16 + c];
  else               v = v2[((size_t)ba*5 + (row-4))*16 + c];
  parts[idx] = v;
}

// ---------------- Wd = sum_d wrel[..., c*16+d, cout]  (the 504MB HBM stream) ----------------
// 4 tiles of 4096 floats per 256-thread block; b128 NT loads, b128 stores.
__global__ void k_wd_reduce(const float* __restrict__ w, float* __restrict__ wd){
  int tile = blockIdx.x*4 + (threadIdx.x>>6);
  int t64  = threadIdx.x & 63;
  int c = t64 >> 2, qg = (t64 & 3)*4;
  const float* src = w + (size_t)tile*4096;
  v4f acc = {0.f,0.f,0.f,0.f};
  #pragma unroll
  for (int d=0; d<16; ++d)
    acc += __builtin_nontemporal_load((const v4f*)(src + c*256 + d*16 + qg));
  *(v4f*)(wd + (size_t)tile*256 + c*16 + qg) = acc;
}

// ---------------- mp (masked neighbor sum) + cg_nl (written in WMMA A-fragment order) ----
// fragment addr: region(lo) + a*16*tau + (K>>2)*64 + ((K>>1)&1)*32 + row*2 + (K&1),
// row = b*(2lo+1)+m, K = slot*256 + c*16 + d.
__global__ void k_mp_cgnl(const float* __restrict__ parts, const float* __restrict__ norms,
                          const float* __restrict__ cg, float* __restrict__ cgnl){
  __shared__ float sMP[9][16];
  __shared__ float sCG[1920];
  int ba = blockIdx.x;                              // 64 blocks, 256 thr
  int b = ba >> 5, a = ba & 31;
  for (int t = threadIdx.x; t < 1920; t += 256) sCG[t] = cg[t];
  if (threadIdx.x < 144){
    int row = threadIdx.x >> 4, c = threadIdx.x & 15;
    float s = 0.f;
    for (int j=0;j<32;++j){
      float cn = (norms[((size_t)b*32 + a)*32 + j] < 0.5f) ? 1.f : 0.f;
      s += cn * parts[(((size_t)b*32 + j)*9 + row)*16 + c];
    }
    sMP[row][c] = s;
  }
  __syncthreads();
  int cc = threadIdx.x >> 4, dd = threadIdx.x & 15;
  for (int lo=0; lo<3; ++lo){
    int nb = LOUT_NBLK_D[lo], nl = 2*lo+1, tau = nb*256;
    size_t base_a = REGC_D[lo] + (size_t)a*16*tau;
    for (int s=0; s<nb; ++s){
      int blk = LOUT_BLK[lo][s];
      int l1 = BLK_L1[blk], l2 = BLK_L2[blk];
      int n1 = 2*l1+1, n2 = 2*l2+1;
      int rb1 = rowb(l1), rb2 = rowb(l2);
      const float* cgb = &sCG[blk*128];
      for (int m=0; m<nl; ++m){
        float v = 0.f;
        for (int x=0; x<n1; ++x){
          float acc = 0.f;
          for (int y=0; y<n2; ++y) acc += cgb[(x*n2+y)*nl + m] * sMP[rb2+y][dd];
          v += acc * sMP[rb1+x][cc];
        }
        int K = s*256 + cc*16 + dd;
        int row = b*nl + m;
        cgnl[ base_a + (size_t)(K>>2)*64 + (size_t)((K>>1)&1)*32 + row*2 + (K&1) ] = v;
      }
    }
  }
}

// ---------------- mixed_nl = cg_nl @ wnl   (WMMA f32 16x16x4) ----------------
template<int LO>
__global__ void k_mixed_nl(const float* __restrict__ cgnl, const float* __restrict__ wnl,
                           float* __restrict__ mnl){
  constexpr int NL  = 2*LO+1;
  constexpr int TAU = (LO==0) ? 768 : 1536;
  constexpr size_t REGC = (LO==0) ? (size_t)0 : (LO==1 ? (size_t)393216 : (size_t)1179648);
  constexpr int RB  = (LO==0) ? 0 : (LO==1 ? 1 : 4);
  int a = blockIdx.x;                               // 32 waves of 32
  const float* ab = cgnl + REGC + (size_t)a*16*TAU;
  const float* wb = wnl + (size_t)a*TAU*16;
  int lane = threadIdx.x, half = lane>>4, l16 = lane&15;

  v8f acc = {0.f,0.f,0.f,0.f,0.f,0.f,0.f,0.f};
  #pragma unroll 4
  for (int k0=0; k0<TAU; k0+=4){
    v2f av = *(const v2f*)(ab + (size_t)(k0>>2)*64 + lane*2);   // coalesced b64, frag order
    int kk = k0 + half*2;
    v2f bv;
    bv.x = wb[(size_t)kk*16 + l16];
    bv.y = wb[(size_t)kk*16 + 16 + l16];
    acc = __builtin_amdgcn_wmma_f32_16x16x4_f32(false, av, false, bv, (short)0, acc, false, false);
  }
  #pragma unroll
  for (int v=0; v<8; ++v){
    int orow = half*8 + v;
    if (orow < 2*NL){
      int obb = orow/NL, om = orow%NL;
      mnl[(((size_t)obb*32 + a)*9 + RB + om)*16 + l16] = acc[v];
    }
  }
}

// ---------------- G written in WMMA A-fragment order ----------------
// region(lo): per i: 512*Kj floats, per j: 16*Kj floats;
// addr: i*512*Kj + j*16*Kj + (K>>2)*64 + ((K>>1)&1)*32 + row*2 + (K&1), row=b*nl+m, K=slot*16+c.
__global__ void k_G(const float* __restrict__ mnl, const float* __restrict__ relpos,
                    const float* __restrict__ cg, float* __restrict__ g0,
                    float* __restrict__ g1, float* __restrict__ g2){
  __shared__ float sCG[1920];
  __shared__ float sM[9][16];
  __shared__ float sY[9];
  int bij = blockIdx.x;                             // 2048 blocks, 256 thr
  int b = bij >> 10, i = (bij >> 5) & 31, j = bij & 31;
  for (int t = threadIdx.x; t < 1920; t += 256) sCG[t] = cg[t];
  if (threadIdx.x < 144)
    sM[threadIdx.x>>4][threadIdx.x&15] = mnl[((size_t)(b*32+i)*9)*16 + threadIdx.x];
  if (threadIdx.x == 0){
    const float* rp = relpos + ((size_t)(b*32+i)*32 + j)*3;
    float x = rp[0], y = rp[1], z = rp[2];
    float r = sqrtf(x*x + y*y + z*z + 1e-6f);
    float ux = x/r, uy = y/r, uz = z/r;
    sY[0] = 0.28209479f;
    sY[1] = 0.48860251f*uy; sY[2] = 0.48860251f*uz; sY[3] = 0.48860251f*ux;
    sY[4] = 1.09254843f*ux*uy; sY[5] = 1.09254843f*uy*uz;
    sY[6] = 0.31539157f*(3.f*uz*uz - 1.f);
    sY[7] = 1.09254843f*ux*uz; sY[8] = 0.54627422f*(ux*ux - uy*uy);
  }
  __syncthreads();
  for (int f = threadIdx.x; f < 816; f += 256){
    int lo, rem;
    if (f < 48)       { lo = 0; rem = f; }
    else if (f < 336) { lo = 1; rem = f - 48; }
    else              { lo = 2; rem = f - 336; }
    int nb = LOUT_NBLK_D[lo], nl = 2*lo+1, Kj = nb*16;
    int m = rem / Kj, K = rem % Kj;
    int s = K >> 4, c = K & 15;
    int blk = LOUT_BLK[lo][s];
    int l1 = BLK_L1[blk], l2 = BLK_L2[blk];
    int n1 = 2*l1+1, n2 = 2*l2+1;
    int rb1 = rowb(l1), yb = rowb(l2);
    const float* cgb = &sCG[blk*128];
    float v = 0.f;
    for (int x=0; x<n1; ++x){
      float acc = 0.f;
      for (int yy=0; yy<n2; ++yy) acc += cgb[(x*n2+yy)*nl + m] * sY[yb+yy];
      v += acc * sM[rb1+x][c];
    }
    float* G = (lo==0) ? g0 : (lo==1 ? g1 : g2);
    int row = b*nl + m;
    G[ (size_t)i*512*Kj + (size_t)j*16*Kj
       + (size_t)(K>>2)*64 + (size_t)((K>>1)&1)*32 + row*2 + (K&1) ] = v;
  }
}

// ---------------- parts_out = sum_j G @ Wd   (WMMA f32 16x16x4, K folds j) ----------------
template<int LO>
__global__ void k_main(const float* __restrict__ gs, const float* __restrict__ wdr,
                       float* __restrict__ pout){
  constexpr int NL = 2*LO+1;
  constexpr int NB = (LO==0) ? 3 : 6;
  constexpr int KJ = NB*16;
  constexpr int RB = (LO==0) ? 0 : (LO==1 ? 1 : 4);
  int i = blockIdx.x;                               // 32 waves of 32
  const float* base  = gs  + (size_t)i*512*KJ;
  const float* wbase = wdr + (size_t)i*32*NB*256;
  int lane = threadIdx.x, half = lane>>4, l16 = lane&15;

  v8f acc = {0.f,0.f,0.f,0.f,0.f,0.f,0.f,0.f};
  for (int j=0; j<32; ++j){
    const float* jb  = base  + (size_t)j*16*KJ;
    const float* bpj = wbase + (size_t)j*NB*256;
    #pragma unroll
    for (int k0=0; k0<KJ; k0+=4){
      v2f av = *(const v2f*)(jb + (k0>>2)*64 + lane*2);         // coalesced b64, frag order
      int kk = k0 + half*2;
      v2f bv;
      bv.x = bpj[kk*16 + l16];
      bv.y = bpj[kk*16 + 16 + l16];
      acc = __builtin_amdgcn_wmma_f32_16x16x4_f32(false, av, false, bv, (short)0, acc, false, false);
    }
  }
  #pragma unroll
  for (int v=0; v<8; ++v){
    int orow = half*8 + v;
    if (orow < 2*NL){
      int obb = orow/NL, om = orow%NL;
      pout[(((size_t)obb*32 + i)*9 + RB + om)*16 + l16] = acc[v];
    }
  }
}

// ---------------- scalar invariants ----------------
__global__ void k_scalars(const float* __restrict__ pout, float* __restrict__ out, int L){
  __shared__ float sP[9][16];
  int ba = blockIdx.x;                              // 64 blocks, 256 thr
  if (threadIdx.x < 144)
    sP[threadIdx.x>>4][threadIdx.x&15] = pout[(size_t)ba*144 + threadIdx.x];
  __syncthreads();
  int b = ba >> 5, a = ba & 31;
  float* ob = out + (((size_t)(L*2 + b))*32 + a)*784;
  if (threadIdx.x < 16) ob[threadIdx.x] = sP[0][threadIdx.x];
  int c = threadIdx.x >> 4, d = threadIdx.x & 15;
  for (int lo=0; lo<3; ++lo){
    int rb = ROWBASE_D[lo], nl = 2*lo+1;
    float s = 0.f;
    for (int m=0; m<nl; ++m) s += sP[rb+m][c] * sP[rb+m][d];
    ob[16 + lo*256 + c*16 + d] = s;
  }
}

// ---------------- launch ----------------
extern "C" void kernel_launch(void* const* d_in, const int* in_sizes, int n_in,
                              void* d_out, int out_size, void* d_ws, size_t ws_size,
                              hipStream_t stream) {
  (void)in_sizes; (void)n_in; (void)out_size; (void)ws_size;
  const float* v0     = (const float*)d_in[0];
  const float* v1     = (const float*)d_in[1];
  const float* v2     = (const float*)d_in[2];
  const float* relpos = (const float*)d_in[3];
  const float* norms  = (const float*)d_in[4];
  float* out = (float*)d_out;
  float* ws  = (float*)d_ws;

  k_cg_init<<<15, 128, 0, stream>>>(ws + OFF_CG);
  k_pack<<<36, 256, 0, stream>>>(v0, v1, v2, ws + OFF_PARTS_A);

  const int    nblkh[3] = {3, 6, 6};
  const size_t loff[3]  = {0, 786432, 2359296};     // per-l offsets for Wd and Gs
  for (int L=0; L<2; ++L)
    for (int l=0; l<3; ++l){
      const float* wrel = (const float*)d_in[6 + L*6 + l*2];
      k_wd_reduce<<<256*nblkh[l], 256, 0, stream>>>(
          wrel, ws + OFF_WD + (size_t)L*WD_LAYER + loff[l]);
    }

  for (int L=0; L<2; ++L){
    float* pin  = ws + (L==0 ? OFF_PARTS_A : OFF_PARTS_B);
    float* pout = ws + (L==0 ? OFF_PARTS_B : OFF_PARTS_A);
    const float* wnl0 = (const float*)d_in[5 + L*6 + 0];
    const float* wnl1 = (const float*)d_in[5 + L*6 + 2];
    const float* wnl2 = (const float*)d_in[5 + L*6 + 4];
    const float* wdL  = ws + OFF_WD + (size_t)L*WD_LAYER;

    k_mp_cgnl<<<64, 256, 0, stream>>>(pin, norms, ws + OFF_CG, ws + OFF_CGNL);
    k_mixed_nl<0><<<32, 32, 0, stream>>>(ws + OFF_CGNL, wnl0, ws + OFF_MNL);
    k_mixed_nl<1><<<32, 32, 0, stream>>>(ws + OFF_CGNL, wnl1, ws + OFF_MNL);
    k_mixed_nl<2><<<32, 32, 0, stream>>>(ws + OFF_CGNL, wnl2, ws + OFF_MNL);
    k_G<<<2048, 256, 0, stream>>>(ws + OFF_MNL, relpos, ws + OFF_CG,
                                  ws + OFF_GS + loff[0], ws + OFF_GS + loff[1],
                                  ws + OFF_GS + loff[2]);
    k_main<0><<<32, 32, 0, stream>>>(ws + OFF_GS + loff[0], wdL + loff[0], pout);
    k_main<1><<<32, 32, 0, stream>>>(ws + OFF_GS + loff[1], wdL + loff[1], pout);
    k_main<2><<<32, 32, 0, stream>>>(ws + OFF_GS + loff[2], wdL + loff[2], pout);
    k_scalars<<<64, 256, 0, stream>>>(pout, out, L);
  }
}